// MoEAdapterLayer_29824252903511
// MI455X (gfx1250) — compile-verified
//
#include <hip/hip_runtime.h>
#include <hip/hip_bf16.h>
#include <math.h>

// MoE adapter layer for MI455X (gfx1250, wave32, WMMA).
//
// Shapes: B=4, L=2048, H=1024, E=8, R=8, TOP_K=2.
// Gate-sparse reformulation: out[b] = x[b] + x[b] @ Acomb[b](1024x16) @ Bcomb[b](16x1024)
// where Acomb = [A_e1 | A_e2] and Bcomb = [g1*B_e1 ; g2*B_e2].
//
// Roofline: ~0.54 GFLOP, ~67 MB mandatory HBM traffic -> memory-bound (~2.9us
// at 23.3 TB/s). 192MB L2 holds all of x, so the stage-2 residual re-read of x
// is an L2 hit. Both GEMM stages run on V_WMMA_F32_16X16X4_F32.
// Stage 1 uses 4 independent accumulators to break the WMMA D->C RAW chain.

#define Bsz 4
#define Lsz 2048
#define Hsz 1024
#define Esz 8
#define Rsz 8

typedef __attribute__((ext_vector_type(2))) float v2f;
typedef __attribute__((ext_vector_type(8))) float v8f;

// Workspace layout (floats):
//   Apack[b][kk(256)][lane(32)][j(2)]           : 4*256*32*2 = 65536 floats
//   Bpack[b][nc(64)][ks(4)][lane(32)][j(2)]     : 4*64*4*32*2 = 65536 floats
// Both are stored in the per-lane order the WMMA B-operand (4x16 f32) wants:
//   lane l supplies (K = kbase + 2*(l>>4) + j, N = l&15).
#define APACK_BATCH (256 * 32 * 2)
#define BPACK_OFF   (Bsz * APACK_BATCH)
#define BPACK_BATCH (64 * 4 * 32 * 2)

// ---------------------------------------------------------------------------
// Kernel 1: router gating + operand pre-swizzle. One block per batch.
// ---------------------------------------------------------------------------
__global__ __launch_bounds__(256) void moe_gate_pack(
    const float* __restrict__ x,        // (B, L, H)
    const float* __restrict__ W_router, // (E, H)
    const float* __restrict__ A_down,   // (E, H, R)
    const float* __restrict__ B_up,     // (E, R, H)
    float* __restrict__ ws) {
  const int b = blockIdx.x;
  const int t = threadIdx.x;

  __shared__ float red[256];
  __shared__ float slog[Esz];
  __shared__ int   sexp[2];
  __shared__ float sgate[2];

  // 1) logits[e] = dot(x[b,0,:], W_router[e,:]) ; 32 lanes per expert.
  {
    const int e = t >> 5;
    const int lane = t & 31;
    const float* xr = x + (size_t)b * Lsz * Hsz;  // row 0 of batch b
    const float* wr = W_router + (size_t)e * Hsz;
    float p = 0.f;
    for (int h = lane; h < Hsz; h += 32) p += xr[h] * wr[h];
    red[t] = p;
  }
  __syncthreads();
  if (t < Esz) {
    float s = 0.f;
    for (int i = 0; i < 32; ++i) s += red[t * 32 + i];
    slog[t] = s;
  }
  __syncthreads();

  // 2) top-2 + softmax over the two kept logits (others are -inf -> weight 0).
  if (t == 0) {
    int i1 = 0;
    float v1 = slog[0];
    for (int e = 1; e < Esz; ++e)
      if (slog[e] > v1) { v1 = slog[e]; i1 = e; }
    int i2 = -1;
    float v2 = -INFINITY;
    for (int e = 0; e < Esz; ++e) {
      if (e == i1) continue;
      if (slog[e] > v2) { v2 = slog[e]; i2 = e; }
    }
    const float e2 = expf(v2 - v1);          // numerically safe softmax
    const float inv = 1.f / (1.f + e2);
    sexp[0] = i1; sexp[1] = i2;
    sgate[0] = inv; sgate[1] = e2 * inv;
  }
  __syncthreads();

  // 3) Pack Acomb into WMMA-B-operand order.
  //    Acomb[h][n] = A_down[sexp[n/8]][h][n%8], h = 4*kk + 2*(l>>4) + j.
  {
    float* ap = ws + (size_t)b * APACK_BATCH;
    for (int it = 0; it < APACK_BATCH / 256; ++it) {
      const int idx = t + 256 * it;
      const int kk = idx >> 6;
      const int l  = (idx >> 1) & 31;
      const int j  = idx & 1;
      const int h  = 4 * kk + 2 * (l >> 4) + j;
      const int n  = l & 15;
      const int e  = sexp[n >> 3];
      const int r  = n & 7;
      ap[idx] = A_down[((size_t)e * Hsz + h) * Rsz + r];
    }
  }

  // 4) Pack gated Bcomb into WMMA-B-operand order.
  //    Bcomb[slot][col] = gate[slot/8] * B_up[sexp[slot/8]][slot%8][col]
  //    slot = 4*ks + 2*(l>>4) + j (rank index), col = nc*16 + (l&15).
  {
    float* bp = ws + BPACK_OFF + (size_t)b * BPACK_BATCH;
    for (int it = 0; it < BPACK_BATCH / 256; ++it) {
      const int idx = t + 256 * it;
      const int nc = idx >> 8;
      const int ks = (idx >> 6) & 3;
      const int l  = (idx >> 1) & 31;
      const int j  = idx & 1;
      const int slot = 4 * ks + 2 * (l >> 4) + j;
      const int col  = nc * 16 + (l & 15);
      const int e = sexp[slot >> 3];
      const int r = slot & 7;
      bp[idx] = sgate[slot >> 3] * B_up[((size_t)e * Rsz + r) * Hsz + col];
    }
  }
}

// ---------------------------------------------------------------------------
// Kernel 2: fused low-rank GEMM + residual. One 16-row L-strip per wave.
// Grid: (L/32, B), block: 64 threads = 2 waves. No divergence (EXEC all 1s).
// ---------------------------------------------------------------------------
__global__ __launch_bounds__(64) void moe_main(
    const float* __restrict__ x,
    const float* __restrict__ ws,
    float* __restrict__ out) {
  const int b    = blockIdx.y;
  const int wave = threadIdx.x >> 5;
  const int lane = threadIdx.x & 31;
  const int strip = blockIdx.x * 2 + wave;
  const int row0  = strip * 16;

  const int r  = lane & 15;        // row within strip for A-operand
  const int h2 = (lane >> 4) * 2;  // K-pair select for A/B operands

  const float* xb = x + ((size_t)b * Lsz + row0) * Hsz;

  // ---- Stage 1: low(16x16) = Xstrip(16x1024) @ Acomb(1024x16) -------------
  // 256 x V_WMMA_F32_16X16X4_F32, K streamed 4 at a time, spread over 4
  // independent accumulators so consecutive WMMAs have no D->C RAW chain.
  v8f acc0 = {}, acc1 = {}, acc2 = {}, acc3 = {};
  const float* apk = ws + (size_t)b * APACK_BATCH + (size_t)lane * 2;
  const float* xrow = xb + (size_t)r * Hsz + h2;
#pragma unroll 2
  for (int kk = 0; kk < 256; kk += 4) {
    v2f a0 = *(const v2f*)(xrow + 4 * (kk + 0));
    v2f a1 = *(const v2f*)(xrow + 4 * (kk + 1));
    v2f a2 = *(const v2f*)(xrow + 4 * (kk + 2));
    v2f a3 = *(const v2f*)(xrow + 4 * (kk + 3));
    v2f b0 = *(const v2f*)(apk + (kk + 0) * 64);
    v2f b1 = *(const v2f*)(apk + (kk + 1) * 64);
    v2f b2 = *(const v2f*)(apk + (kk + 2) * 64);
    v2f b3 = *(const v2f*)(apk + (kk + 3) * 64);
    acc0 = __builtin_amdgcn_wmma_f32_16x16x4_f32(false, a0, false, b0, (short)0, acc0, false, false);
    acc1 = __builtin_amdgcn_wmma_f32_16x16x4_f32(false, a1, false, b1, (short)0, acc1, false, false);
    acc2 = __builtin_amdgcn_wmma_f32_16x16x4_f32(false, a2, false, b2, (short)0, acc2, false, false);
    acc3 = __builtin_amdgcn_wmma_f32_16x16x4_f32(false, a3, false, b3, (short)0, acc3, false, false);
  }
  v8f low = (acc0 + acc1) + (acc2 + acc3);  // same D layout -> elementwise sum

  // ---- Re-shape low from D-layout to A-operand layout via padded LDS ------
  __shared__ __align__(16) float lds_low[2][16 * 18];  // stride 18: bank-safe
  float* Lw = &lds_low[wave][0];
#pragma unroll
  for (int v = 0; v < 8; ++v)
    Lw[(v + 8 * (lane >> 4)) * 18 + (lane & 15)] = low[v];

  v2f a2op[4];  // hoisted stage-2 A operands (same-wave LDS ops are in-order)
#pragma unroll
  for (int ks = 0; ks < 4; ++ks)
    a2op[ks] = *(const v2f*)(&Lw[r * 18 + 4 * ks + h2]);

  // ---- Stage 2: out = x + low(16x16) @ Bcomb(16x1024) ---------------------
  const float* bpk = ws + BPACK_OFF + (size_t)b * BPACK_BATCH + (size_t)lane * 2;
  float* outp = out + ((size_t)b * Lsz + row0) * Hsz;
  for (int nc = 0; nc < 64; ++nc) {
    v8f y = {};
#pragma unroll
    for (int ks = 0; ks < 4; ++ks) {
      v2f bo = *(const v2f*)(bpk + nc * 256 + ks * 64);
      y = __builtin_amdgcn_wmma_f32_16x16x4_f32(
          false, a2op[ks], false, bo, (short)0, y, false, false);
    }
    // D layout: lane l, VGPR v -> (M = v + 8*(l>>4), N = nc*16 + (l&15)).
    const size_t cbase = (size_t)nc * 16 + (lane & 15);
#pragma unroll
    for (int v = 0; v < 8; ++v) {
      const size_t o = (size_t)(v + 8 * (lane >> 4)) * Hsz + cbase;
      outp[o] = xb[o] + y[v];  // residual re-read of x hits L2 (192MB >> 33MB)
    }
  }
}

// ---------------------------------------------------------------------------
extern "C" void kernel_launch(void* const* d_in, const int* in_sizes, int n_in,
                              void* d_out, int out_size, void* d_ws, size_t ws_size,
                              hipStream_t stream) {
  const float* x        = (const float*)d_in[0];
  const float* W_router = (const float*)d_in[1];
  const float* A_down   = (const float*)d_in[2];
  const float* B_up     = (const float*)d_in[3];
  float* out = (float*)d_out;
  float* ws  = (float*)d_ws;  // needs 131072 floats = 512 KB

  moe_gate_pack<<<dim3(Bsz), dim3(256), 0, stream>>>(x, W_router, A_down, B_up, ws);
  moe_main<<<dim3(Lsz / 32, Bsz), dim3(64), 0, stream>>>(x, ws, out);
}